// YibeiSynthSemantic_74156905333422
// MI455X (gfx1250) — compile-verified
//
#include <hip/hip_runtime.h>
#include <hip/hip_bf16.h>

// Elementwise LUT-gather "vessel painting" kernel for MI455X (gfx1250).
// Memory-roofline bound: ~402 MB traffic -> ~17us at 23.3 TB/s.
//  - vec4 (b128) non-temporal streaming of labels/parenchyma/outputs
//  - 256-entry {scale, class} LUT fused into LDS (one ds_load_b64 per voxel)
//  - LUT tables staged into LDS via the Tensor Data Mover (tensor_load_to_lds
//    + s_wait_tensorcnt) when the builtin is available; global-load fallback
//  - global_prefetch_b8 on the next grid-stride tile

typedef __attribute__((ext_vector_type(4))) float        v4f;
typedef __attribute__((ext_vector_type(4))) int          v4i;
typedef __attribute__((ext_vector_type(2))) float        v2f;
typedef __attribute__((ext_vector_type(4))) unsigned int u32x4;
typedef __attribute__((ext_vector_type(8))) int          i32x8;
typedef __attribute__((ext_vector_type(4))) int          i32x4;

#define NIDS 256

#if __has_builtin(__builtin_amdgcn_tensor_load_to_lds) && \
    __has_builtin(__builtin_amdgcn_s_wait_tensorcnt)
#define USE_TDM 1
#else
#define USE_TDM 0
#endif

#if USE_TDM
// TDM 1-D tile load: 256 x 4-byte elements from `gptr` into LDS byte offset
// `lds_off`. D# per CDNA5 ISA §8.3/8.4: group0 = {count=1, lds_addr,
// global_addr[56:0], type=2}; group1 = {wg_mask=0, data_size=2 (4B),
// tensor_dim0=256, tensor_dim1=1, tile_dim0=256, tile_dim1/2 unused,
// tensor_dim0_stride=256}; groups 2/3 zero (dims 2+ unused).
__device__ __forceinline__ void tdm_load_1kb(const void* gptr, unsigned lds_off) {
    unsigned long long ga = (unsigned long long)(size_t)gptr;
    u32x4 g0;
    g0.x = 1u;                                               // count=1, user desc
    g0.y = lds_off;                                          // LDS byte address
    g0.z = (unsigned)(ga & 0xFFFFFFFFu);                     // global_addr[31:0]
    g0.w = (unsigned)((ga >> 32) & 0x01FFFFFFu) | (2u << 30);// [56:32] | type=2
    i32x8 g1;
    g1[0] = 0x00020000;          // workgroup_mask=0, data_size=2 -> 4 bytes
    g1[1] = (int)(256u << 16);   // atomic_barrier_addr=0; tensor_dim0.lo16=256
    g1[2] = (int)(1u << 16);     // tensor_dim0.hi16=0;    tensor_dim1.lo16=1
    g1[3] = (int)(256u << 16);   // tensor_dim1.hi16=0;    tile_dim0=256
    g1[4] = 0;                   // tile_dim1=0 (unused), tile_dim2=0
    g1[5] = 256;                 // tensor_dim0_stride = 256 elements
    g1[6] = 0;                   // stride hi / tensor_dim1_stride = 0
    g1[7] = 0;
    i32x4 z4 = (i32x4)0;
#if __clang_major__ >= 23
    i32x8 z8 = (i32x8)0;
    __builtin_amdgcn_tensor_load_to_lds(g0, g1, z4, z4, z8, 0);
#else
    __builtin_amdgcn_tensor_load_to_lds(g0, g1, z4, z4, 0);
#endif
}
#endif

__global__ __launch_bounds__(256) void vessel_paint_kernel(
    const v4i*   __restrict__ labels4,      // [n4] int32 x4 vessel labels
    const float* __restrict__ id_intensity, // [256]
    const int*   __restrict__ id_is_dark,   // [256]
    const v4f*   __restrict__ par4,         // [n4] f32 x4 parenchyma
    float*       __restrict__ out,          // [4*vox]: out, bg, dark, bright
    int n4, long long vox)
{
    // Single shared aggregate -> member offsets are the LDS byte offsets
    // (base of the kernel's static LDS allocation).
    __shared__ struct {
        float rawI[NIDS];   // byte offset 0     (TDM target)
        int   rawD[NIDS];   // byte offset 1024  (TDM target)
        v2f   lut[NIDS];    // byte offset 2048  (fused {scale, class})
    } smem;

    const int t = threadIdx.x;   // blockDim.x == 256 -> t in [0,255]

#if USE_TDM
    if (t < 32) {   // wave 0 issues the DMAs; other waves branch over
        tdm_load_1kb((const void*)id_intensity, 0u);
        tdm_load_1kb((const void*)id_is_dark, 1024u);
        __builtin_amdgcn_s_wait_tensorcnt(0);
    }
    __syncthreads();
    const float s_raw = smem.rawI[t];
    const int   d_raw = smem.rawD[t];
#else
    const float s_raw = id_intensity[t];
    const int   d_raw = id_is_dark[t];
#endif
    // Branch-free fused LUT build: id 0 -> {1.0, class 0 (background)};
    // else {intensity, 1=dark / 2=bright}.
    v2f e;
    e.x = (t == 0) ? 1.0f : s_raw;
    e.y = (t == 0) ? 0.0f : ((d_raw == 1) ? 1.0f : 2.0f);
    smem.lut[t] = e;
    __syncthreads();

    v4f* __restrict__ o_out = (v4f*)(out);
    v4f* __restrict__ o_bg  = (v4f*)(out + vox);
    v4f* __restrict__ o_dk  = (v4f*)(out + 2 * vox);
    v4f* __restrict__ o_br  = (v4f*)(out + 3 * vox);

    const int stride = (int)(gridDim.x * blockDim.x);
    for (int i = (int)(blockIdx.x * blockDim.x) + t; i < n4; i += stride) {
        // Prefetch next grid-stride tile (speculative).
        int ip = i + stride;
        if (ip >= n4) ip = i;
        __builtin_prefetch((const void*)&labels4[ip], 0, 0);
        __builtin_prefetch((const void*)&par4[ip], 0, 0);

        // Streaming reads: touched exactly once -> non-temporal b128.
        v4i id = __builtin_nontemporal_load(&labels4[i]);
        v4f p  = __builtin_nontemporal_load(&par4[i]);

        // One ds_load_b64 fused-LUT gather per voxel.
        v2f e0 = smem.lut[id.x & (NIDS - 1)];
        v2f e1 = smem.lut[id.y & (NIDS - 1)];
        v2f e2 = smem.lut[id.z & (NIDS - 1)];
        v2f e3 = smem.lut[id.w & (NIDS - 1)];

        // out = parenchyma * scale (scale==1.0 for background -> exact).
        v4f o;
        o.x = p.x * e0.x;
        o.y = p.y * e1.x;
        o.z = p.z * e2.x;
        o.w = p.w * e3.x;

        // One-hot channels from the class code.
        v4f bg, dk, br;
        bg.x = (e0.y == 0.0f) ? 1.0f : 0.0f;
        bg.y = (e1.y == 0.0f) ? 1.0f : 0.0f;
        bg.z = (e2.y == 0.0f) ? 1.0f : 0.0f;
        bg.w = (e3.y == 0.0f) ? 1.0f : 0.0f;
        dk.x = (e0.y == 1.0f) ? 1.0f : 0.0f;
        dk.y = (e1.y == 1.0f) ? 1.0f : 0.0f;
        dk.z = (e2.y == 1.0f) ? 1.0f : 0.0f;
        dk.w = (e3.y == 1.0f) ? 1.0f : 0.0f;
        br.x = (e0.y == 2.0f) ? 1.0f : 0.0f;
        br.y = (e1.y == 2.0f) ? 1.0f : 0.0f;
        br.z = (e2.y == 2.0f) ? 1.0f : 0.0f;
        br.w = (e3.y == 2.0f) ? 1.0f : 0.0f;

        // Streaming writes: 268 MB total (> 192 MB L2) -> non-temporal b128.
        __builtin_nontemporal_store(o,  &o_out[i]);
        __builtin_nontemporal_store(bg, &o_bg[i]);
        __builtin_nontemporal_store(dk, &o_dk[i]);
        __builtin_nontemporal_store(br, &o_br[i]);
    }
}

extern "C" void kernel_launch(void* const* d_in, const int* in_sizes, int n_in,
                              void* d_out, int out_size, void* d_ws, size_t ws_size,
                              hipStream_t stream) {
    // setup_inputs order: vessel_labels [vox] i32, id_intensity [256] f32,
    //                     id_is_dark [256] i32, parenchyma [vox] f32.
    const v4i*   labels4      = (const v4i*)d_in[0];
    const float* id_intensity = (const float*)d_in[1];
    const int*   id_is_dark   = (const int*)d_in[2];
    const v4f*   par4         = (const v4f*)d_in[3];
    float*       out          = (float*)d_out;

    const long long vox = (long long)in_sizes[0];  // 256^3
    const int n4 = (int)(vox / 4);

    dim3 block(256);
    dim3 grid(4096);  // 1M threads, ~4 vec4 iterations each; 8 wave32 / block
    vessel_paint_kernel<<<grid, block, 0, stream>>>(
        labels4, id_intensity, id_is_dark, par4, out, n4, vox);
}